// GATNet_25340307046789
// MI455X (gfx1250) — compile-verified
//
#include <hip/hip_runtime.h>
#include <hip/hip_bf16.h>

typedef float v2f __attribute__((ext_vector_type(2)));
typedef float v8f __attribute__((ext_vector_type(8)));

#define NN 50000
#define NE 800000
#define ETOT 850000   // NE + NN self loops
#define NG 512

// ---- ordered-uint mapping for float atomic max ----
__device__ __forceinline__ unsigned f2ord(float f) {
  unsigned u = __float_as_uint(f);
  return (u & 0x80000000u) ? ~u : (u | 0x80000000u);
}
__device__ __forceinline__ float ord2f(unsigned u) {
  return (u & 0x80000000u) ? __uint_as_float(u & 0x7fffffffu)
                           : __uint_as_float(~u);
}

__global__ void k_fill(float* p, float v, long long n) {
  long long i = (long long)blockIdx.x * blockDim.x + threadIdx.x;
  if (i < n) p[i] = v;
}

// ---------------- Layer-1 GEMM: [50000,3] @ [3,256] via one WMMA f32 16x16x4 per tile
// block = 512 threads (16 waves); wave w handles N-tile w (16 cols); grid.x = M tiles (3125)
__global__ void k_gemm1(const float* __restrict__ X, const float* __restrict__ W1,
                        float* __restrict__ H1) {
  int lane = threadIdx.x & 31;
  int wave = threadIdx.x >> 5;           // 0..15
  int m0   = blockIdx.x * 16;
  int hi   = lane >> 4;                  // 0 -> K=0,1 ; 1 -> K=2,3 (K=3 is zero pad)
  int mrow = m0 + (lane & 15);
  int ncol = wave * 16 + (lane & 15);
  v2f a, b;
  if (hi == 0) {
    a.x = X[mrow * 3 + 0]; a.y = X[mrow * 3 + 1];
    b.x = W1[0 * 256 + ncol]; b.y = W1[1 * 256 + ncol];
  } else {
    a.x = X[mrow * 3 + 2]; a.y = 0.f;
    b.x = W1[2 * 256 + ncol]; b.y = 0.f;
  }
  v8f acc = {0.f,0.f,0.f,0.f,0.f,0.f,0.f,0.f};
  acc = __builtin_amdgcn_wmma_f32_16x16x4_f32(false, a, false, b, (short)0, acc,
                                              false, false);
  int mbase = m0 + (hi << 3);
#pragma unroll
  for (int r = 0; r < 8; ++r)
    H1[(size_t)(mbase + r) * 256 + ncol] = acc[r];
}

// ---------------- Layer-2 GEMM: [50000,256] @ [256,128] via WMMA f32 16x16x4, K loop
// block = 256 threads (8 waves); wave w -> N-tile w (N=128 = 8 tiles); grid.x = 3125
__global__ void k_gemm2(const float* __restrict__ A, const float* __restrict__ B,
                        float* __restrict__ C) {
  int lane = threadIdx.x & 31;
  int wave = threadIdx.x >> 5;           // 0..7
  int m0   = blockIdx.x * 16;
  int n0   = wave * 16;
  int hi   = lane >> 4;
  int kofs = hi * 2;
  int mrow = m0 + (lane & 15);
  int ncol = n0 + (lane & 15);
  const float* ap = A + (size_t)mrow * 256 + kofs;
  const float* bp = B + (size_t)kofs * 128 + ncol;
  v8f acc = {0.f,0.f,0.f,0.f,0.f,0.f,0.f,0.f};
#pragma unroll 4
  for (int k0 = 0; k0 < 256; k0 += 4) {
    v2f a; a.x = ap[k0];       a.y = ap[k0 + 1];
    v2f b; b.x = bp[k0 * 128]; b.y = bp[k0 * 128 + 128];
    acc = __builtin_amdgcn_wmma_f32_16x16x4_f32(false, a, false, b, (short)0, acc,
                                                false, false);
  }
  int mbase = m0 + (hi << 3);
  float* cp = C + (size_t)mbase * 128 + ncol;
#pragma unroll
  for (int r = 0; r < 8; ++r) cp[r * 128] = acc[r];
}

// ---------------- attention dot products: a_src[n,h] = <h[n,h,:], att_src[h,:]>
template <int H, int C>
__global__ void k_att(const float* __restrict__ h, const float* __restrict__ att_s,
                      const float* __restrict__ att_d, float* __restrict__ as,
                      float* __restrict__ ad) {
  int tid = blockIdx.x * blockDim.x + threadIdx.x;
  if (tid >= NN * H) return;
  int n = tid / H, hd = tid - n * H;
  constexpr int CH = C / H;
  const float* hp = h + (size_t)n * C + hd * CH;
  const float* sp = att_s + hd * CH;
  const float* dp = att_d + hd * CH;
  float a = 0.f, b = 0.f;
#pragma unroll 8
  for (int c = 0; c < CH; ++c) { a += hp[c] * sp[c]; b += hp[c] * dp[c]; }
  as[tid] = a;
  ad[tid] = b;
}

// ---------------- pass 1: e = leakyrelu(asrc[src]+adst[dst]); segment max over dst
template <int H>
__global__ void k_edge_max(const int* __restrict__ ei, const float* __restrict__ asrc,
                           const float* __restrict__ adst, float* __restrict__ eraw,
                           unsigned* __restrict__ emax) {
  int tid = blockIdx.x * blockDim.x + threadIdx.x;
  if (tid >= ETOT * H) return;
  int eid = tid / H, hd = tid - eid * H;
  int src, dst;
  if (eid < NE) { src = ei[eid]; dst = ei[NE + eid]; }
  else          { src = dst = eid - NE; }
  float e = asrc[src * H + hd] + adst[dst * H + hd];
  e = e > 0.f ? e : 0.2f * e;
  eraw[tid] = e;
  atomicMax(&emax[dst * H + hd], f2ord(e));
}

// ---------------- pass 2: ee = exp(e - emax[dst]); segment sum -> denom
template <int H>
__global__ void k_edge_exp(const int* __restrict__ ei, const unsigned* __restrict__ emax,
                           float* __restrict__ ee, float* __restrict__ denom) {
  int tid = blockIdx.x * blockDim.x + threadIdx.x;
  if (tid >= ETOT * H) return;
  int eid = tid / H, hd = tid - eid * H;
  int dst = (eid < NE) ? ei[NE + eid] : (eid - NE);
  float e = ee[tid];
  float v = __expf(e - ord2f(emax[dst * H + hd]));
  ee[tid] = v;
  atomicAdd(&denom[dst * H + hd], v);
}

// ---------------- pass 3: out[dst] += alpha * h[src]  (C/4 threads per edge, float4)
template <int H, int C>
__global__ void k_edge_aggr(const int* __restrict__ ei, const float* __restrict__ ee,
                            const float* __restrict__ denom, const float* __restrict__ h,
                            float* __restrict__ outp) {
  constexpr int TPE = C / 4;
  int tid = blockIdx.x * blockDim.x + threadIdx.x;
  if (tid >= ETOT * TPE) return;
  int eid = tid / TPE;
  int t   = tid - eid * TPE;
  int c0  = t * 4;
  constexpr int CH = C / H;
  int hd = c0 / CH;
  int src, dst;
  if (eid < NE) { src = ei[eid]; dst = ei[NE + eid]; }
  else          { src = dst = eid - NE; }
  float alpha = ee[eid * H + hd] / (denom[dst * H + hd] + 1e-16f);
  float4 hv = *(const float4*)(h + (size_t)src * C + c0);
  float* op = outp + (size_t)dst * C + c0;
  atomicAdd(op + 0, alpha * hv.x);
  atomicAdd(op + 1, alpha * hv.y);
  atomicAdd(op + 2, alpha * hv.z);
  atomicAdd(op + 3, alpha * hv.w);
}

template <int C>
__global__ void k_bias_relu(float* __restrict__ o, const float* __restrict__ bias,
                            int total) {
  int tid = blockIdx.x * blockDim.x + threadIdx.x;
  if (tid >= total) return;
  float v = o[tid] + bias[tid & (C - 1)];
  o[tid] = v > 0.f ? v : 0.f;
}

__global__ void k_pool(const float* __restrict__ h, const int* __restrict__ batch,
                       float* __restrict__ pool, float* __restrict__ cnt) {
  int tid = blockIdx.x * blockDim.x + threadIdx.x;
  if (tid >= NN * 128) return;
  int n = tid >> 7, c = tid & 127;
  int g = batch[n];
  atomicAdd(&pool[g * 128 + c], h[tid]);
  if (c == 0) atomicAdd(&cnt[g], 1.0f);
}

__global__ void k_fc(const float* __restrict__ pool, const float* __restrict__ cnt,
                     const float* __restrict__ fcW, const float* __restrict__ fcb,
                     float* __restrict__ out) {
  int tid = blockIdx.x * blockDim.x + threadIdx.x;
  if (tid >= NG * 2) return;
  int g = tid >> 1, j = tid & 1;
  float inv = 1.0f / fmaxf(cnt[g], 1.0f);
  float s = 0.f;
#pragma unroll 8
  for (int c = 0; c < 128; ++c) s += pool[g * 128 + c] * inv * fcW[c * 2 + j];
  out[tid] = s + fcb[j];
}

static inline int gdiv(long long n, int b) { return (int)((n + b - 1) / b); }

extern "C" void kernel_launch(void* const* d_in, const int* in_sizes, int n_in,
                              void* d_out, int out_size, void* d_ws, size_t ws_size,
                              hipStream_t stream) {
  const float* x    = (const float*)d_in[0];
  const int*   ei   = (const int*)d_in[1];     // [2, 800000]
  const int*   batch= (const int*)d_in[2];
  const float* W1   = (const float*)d_in[3];
  const float* as1w = (const float*)d_in[4];
  const float* ad1w = (const float*)d_in[5];
  const float* b1   = (const float*)d_in[6];
  const float* W2   = (const float*)d_in[7];
  const float* as2w = (const float*)d_in[8];
  const float* ad2w = (const float*)d_in[9];
  const float* b2   = (const float*)d_in[10];
  const float* fcW  = (const float*)d_in[11];
  const float* fcb  = (const float*)d_in[12];

  float* ws = (float*)d_ws;
  float*    h1     = ws;                       // NN*256
  float*    out1   = h1    + 12800000;         // NN*256 (becomes h1post)
  float*    asrc1  = out1  + 12800000;         // NN*4
  float*    adst1  = asrc1 + 200000;           // NN*4
  unsigned* emax1  = (unsigned*)(adst1 + 200000); // NN*4
  float*    denom1 = (float*)(emax1 + 200000); // NN*4
  float*    ee1    = denom1 + 200000;          // ETOT*4
  float*    h2     = ee1   + 3400000;          // NN*128
  float*    out2   = h2    + 6400000;          // NN*128 (becomes h2post)
  float*    asrc2  = out2  + 6400000;          // NN
  float*    adst2  = asrc2 + 50000;            // NN
  unsigned* emax2  = (unsigned*)(adst2 + 50000); // NN
  float*    denom2 = (float*)(emax2 + 50000);  // NN
  float*    ee2    = denom2 + 50000;           // ETOT
  float*    poolb  = ee2   + 850000;           // NG*128
  float*    cnt    = poolb + 65536;            // NG

  const int B = 256;

  // ---- layer 1 ----
  k_gemm1<<<3125, 512, 0, stream>>>(x, W1, h1);
  k_att<4, 256><<<gdiv((long long)NN * 4, B), B, 0, stream>>>(h1, as1w, ad1w, asrc1, adst1);
  k_fill<<<gdiv(200000, B), B, 0, stream>>>((float*)emax1, 0.f, 200000);   // ord(-inf) floor
  k_fill<<<gdiv(200000, B), B, 0, stream>>>(denom1, 0.f, 200000);
  k_fill<<<gdiv(12800000, B), B, 0, stream>>>(out1, 0.f, 12800000);
  k_edge_max<4><<<gdiv((long long)ETOT * 4, B), B, 0, stream>>>(ei, asrc1, adst1, ee1, emax1);
  k_edge_exp<4><<<gdiv((long long)ETOT * 4, B), B, 0, stream>>>(ei, emax1, ee1, denom1);
  k_edge_aggr<4, 256><<<gdiv((long long)ETOT * 64, B), B, 0, stream>>>(ei, ee1, denom1, h1, out1);
  k_bias_relu<256><<<gdiv(12800000, B), B, 0, stream>>>(out1, b1, 12800000);

  // ---- layer 2 ----
  k_gemm2<<<3125, 256, 0, stream>>>(out1, W2, h2);
  k_att<1, 128><<<gdiv(NN, B), B, 0, stream>>>(h2, as2w, ad2w, asrc2, adst2);
  k_fill<<<gdiv(50000, B), B, 0, stream>>>((float*)emax2, 0.f, 50000);
  k_fill<<<gdiv(50000, B), B, 0, stream>>>(denom2, 0.f, 50000);
  k_fill<<<gdiv(6400000, B), B, 0, stream>>>(out2, 0.f, 6400000);
  k_edge_max<1><<<gdiv(ETOT, B), B, 0, stream>>>(ei, asrc2, adst2, ee2, emax2);
  k_edge_exp<1><<<gdiv(ETOT, B), B, 0, stream>>>(ei, emax2, ee2, denom2);
  k_edge_aggr<1, 128><<<gdiv((long long)ETOT * 32, B), B, 0, stream>>>(ei, ee2, denom2, h2, out2);
  k_bias_relu<128><<<gdiv(6400000, B), B, 0, stream>>>(out2, b2, 6400000);

  // ---- pool + FC ----
  k_fill<<<gdiv(65536, B), B, 0, stream>>>(poolb, 0.f, 65536);
  k_fill<<<gdiv(512, B), B, 0, stream>>>(cnt, 0.f, 512);
  k_pool<<<gdiv((long long)NN * 128, B), B, 0, stream>>>(out2, batch, poolb, cnt);
  k_fc<<<gdiv(NG * 2, B), B, 0, stream>>>(poolb, cnt, fcW, fcb, (float*)d_out);
}